// FoldNd_14559939133583
// MI455X (gfx1250) — compile-verified
//
#include <hip/hip_runtime.h>

// FoldNd (col2im): B=8, C=32, H=W=256, K=3, stride=1, pad=0, dil=1
// LH=LW=254, L=64516, K2=9.
// Gather formulation: out[b,c,h,w] = sum_{kh,kw} x[b, c*9+kh*3+kw, (h-kh)*254 + (w-kw)]
// valid when 0 <= h-kh < 254 and 0 <= w-kw < 254.
//
// Branchless version: clamp (oh,ow) to a valid in-slice element, load
// unconditionally (all 9 loads issued back-to-back -> MLP=9 per wave,
// single loadcnt wait), then zero-mask invalid taps with a select.

#define FB   8
#define FC   32
#define FH   256
#define FW   256
#define FLH  254
#define FLW  254
#define FL   (FLH * FLW)   // 64516

__global__ __launch_bounds__(256) void fold_gather_kernel(
    const float* __restrict__ x, float* __restrict__ out) {
  const int w   = threadIdx.x;           // 0..255
  const int bch = blockIdx.x;            // (b*C + c)*H + h, 0..65535
  const int h   = bch & (FH - 1);
  const int bc  = bch >> 8;              // b*C + c

  // Base of the 9 contiguous channel slices for this (b,c): x is (B, C*9, L)
  const float* __restrict__ xs = x + (size_t)bc * 9u * (size_t)FL;

  float v[9];
  bool  ok[9];
#pragma unroll
  for (int kh = 0; kh < 3; ++kh) {
    const int  oh  = h - kh;
    const bool okh = (unsigned)oh < (unsigned)FLH;   // wave-uniform
    const int  ohc = okh ? oh : 0;                   // clamped, always valid
    const float* __restrict__ rowbase =
        xs + (size_t)(kh * 3) * (size_t)FL + (size_t)ohc * (size_t)FLW;
#pragma unroll
    for (int kw = 0; kw < 3; ++kw) {
      const int  ow  = w - kw;
      const bool okw = (unsigned)ow < (unsigned)FLW; // diverges on <=2 lanes
      const int  owc = okw ? ow : 0;                 // clamped, always valid
      const int  k   = kh * 3 + kw;
      ok[k] = okh && okw;
      // Unconditional, read-once, non-temporal load (gfx1250 TH=NT).
      v[k] = __builtin_nontemporal_load(rowbase + (size_t)kw * (size_t)FL + owc);
    }
  }

  float acc = 0.0f;
#pragma unroll
  for (int k = 0; k < 9; ++k)
    acc += ok[k] ? v[k] : 0.0f;

  // Streaming write-once output: non-temporal store.
  __builtin_nontemporal_store(acc, out + (size_t)bch * (size_t)FW + w);
}

extern "C" void kernel_launch(void* const* d_in, const int* in_sizes, int n_in,
                              void* d_out, int out_size, void* d_ws, size_t ws_size,
                              hipStream_t stream) {
  (void)in_sizes; (void)n_in; (void)out_size; (void)d_ws; (void)ws_size;
  const float* x = (const float*)d_in[0];
  float* out = (float*)d_out;

  // One block per output row (b,c,h): 8*32*256 = 65536 blocks, 256 threads
  // (8 wave32s) each; thread = one output pixel along w.
  fold_gather_kernel<<<dim3(FB * FC * FH), dim3(FW), 0, stream>>>(x, out);
}